// BlurPool_48747878810158
// MI455X (gfx1250) — compile-verified
//
#include <hip/hip_runtime.h>
#include <stdint.h>

// BlurPool3D: depthwise separable [1,4,6,4,1]/16 blur per axis, stride 2,
// edge-replicate pad 2.  x:[2,32,96,96,96] f32 -> out:[2,32,48,48,48] f32.
// Memory-bound (255 MB @ 23.3 TB/s ~ 11 us): fully fused through big LDS
// tiles streamed in with CDNA5 async global->LDS b64 copies.

#define N_SLICES   64          // 2 * 32 (batch x channels)
#define IN_DIM     96
#define OUT_DIM    48

// Output tile per workgroup
#define TW 16
#define TH 16
#define TD 8
// Input tile (stride-2 conv, 5 taps): 2*T + 3
#define IW (2*TW + 3)          // 35 (logical)
#define IH (2*TH + 3)          // 35
#define ID (2*TD + 3)          // 19
#define IWP 36                 // padded LDS row stride: keeps b64 slots 8B-aligned

#define NROWS  (ID*IH)         // 665 rows
#define NRAWP  (NROWS*IWP)     // 23940 floats (95.8 KB)
#define NTMPW  (ID*IH*TW)      // 10640 floats (42.6 KB)
#define NTMPH  (ID*TH*TW)      //  4864 floats (aliases raw storage)

#define NTHREADS 256
#define LDS_BYTES ((NRAWP + NTMPW) * (int)sizeof(float))   // 138,320 B (< 320 KB WGP LDS)

__global__ __launch_bounds__(NTHREADS)
void blurpool3d_kernel(const float* __restrict__ x, float* __restrict__ out)
{
    extern __shared__ float lds[];
    float* raw  = lds;               // [ID*IH][IWP] input tile (row-padded)
    float* tmpW = lds + NRAWP;       // [ID][IH][TW] after W pass
    float* tmpH = lds;               // [ID][TH][TW] after H pass (reuses raw)

    const int tid   = threadIdx.x;
    const int wt    = blockIdx.x;                    // 0..2
    const int ht    = blockIdx.y;                    // 0..2
    const int zt    = blockIdx.z;                    // 0..(6*64-1)
    const int dt    = zt % (OUT_DIM / TD);           // 0..5
    const int slice = zt / (OUT_DIM / TD);           // 0..63

    const int ow0 = wt * TW, oh0 = ht * TH, od0 = dt * TD;
    const int iw0 = 2*ow0 - 2, ih0 = 2*oh0 - 2, id0 = 2*od0 - 2;

    // Valid (in-bounds) column range of each input row; always starts even
    // (gw_lo = 0 / 30 / 62) so b64 transfers stay 8B-aligned end to end.
    const int gw_lo  = iw0 < 0 ? 0 : iw0;
    const int wstart = gw_lo - iw0;                  // 2 for wt==0, else 0
    const int npairs = (wt == 1) ? 18 : 17;          // b64 slots per row

    const float* __restrict__ xs =
        x + (size_t)slice * (IN_DIM * IN_DIM * IN_DIM);

    // ---- Stage 1: async b64 global -> LDS of the in-bounds column range ----
    // H/D edge clamp folded into the (row-contiguous) address; W edge
    // replication done later as an LDS fixup.  Tracked by ASYNCcnt.
    for (int t = tid; t < NROWS * 18; t += NTHREADS) {
        int p   = t % 18;
        int row = t / 18;
        if (p < npairs) {
            int h = row % IH;
            int d = row / IH;
            int gh = ih0 + h; gh = gh < 0 ? 0 : (gh > IN_DIM-1 ? IN_DIM-1 : gh);
            int gd = id0 + d; gd = gd < 0 ? 0 : (gd > IN_DIM-1 ? IN_DIM-1 : gd);
            size_t gidx = ((size_t)gd * IN_DIM + gh) * IN_DIM + (gw_lo + 2*p);

            unsigned long long ga = (unsigned long long)(uintptr_t)(xs + gidx);
            // Generic LDS pointer: low 32 bits are the LDS byte offset.
            unsigned la =
                (unsigned)(uintptr_t)(raw + (row * IWP + wstart + 2*p));

            asm volatile("global_load_async_to_lds_b64 %0, %1, off"
                         :: "v"(la), "v"(ga)
                         : "memory");
        }
    }
    // Wait for this wave's async transfers, then make all waves' LDS visible.
    asm volatile("s_wait_asynccnt 0" ::: "memory");
    __syncthreads();

    // ---- Stage 1b: W-edge replicate fixup in LDS (wt is block-uniform) ----
    if (wt == 0) {
        for (int r = tid; r < NROWS; r += NTHREADS) {
            float v = raw[r * IWP + 2];      // column gw==0
            raw[r * IWP + 0] = v;
            raw[r * IWP + 1] = v;
        }
        __syncthreads();
    } else if (wt == 2) {
        for (int r = tid; r < NROWS; r += NTHREADS) {
            raw[r * IWP + 34] = raw[r * IWP + 33];   // replicate gw==95
        }
        __syncthreads();
    }

    // Exact separable weights: [1,4,6,4,1]/16 per axis (all exact in f32).
    const float w0 = 0.0625f, w1 = 0.25f, w2 = 0.375f;

    // ---- Stage 2: W pass (stride 2) raw -> tmpW ----
    for (int i = tid; i < NTMPW; i += NTHREADS) {
        int wo = i % TW;
        int t2 = i / TW;
        int h  = t2 % IH;
        int d  = t2 / IH;
        const float* r = &raw[(d*IH + h)*IWP + 2*wo];
        tmpW[i] = w0*(r[0] + r[4]) + w1*(r[1] + r[3]) + w2*r[2];
    }
    __syncthreads();

    // ---- Stage 3: H pass (stride 2) tmpW -> tmpH (reuses raw storage) ----
    for (int i = tid; i < NTMPH; i += NTHREADS) {
        int wo = i % TW;
        int t2 = i / TW;
        int ho = t2 % TH;
        int d  = t2 / TH;
        const float* r = &tmpW[(d*IH + 2*ho)*TW + wo];
        tmpH[i] = w0*(r[0*TW] + r[4*TW]) + w1*(r[1*TW] + r[3*TW]) + w2*r[2*TW];
    }
    __syncthreads();

    // ---- Stage 4: D pass (stride 2) tmpH -> global store (coalesced) ----
    float* __restrict__ os =
        out + (size_t)slice * (OUT_DIM * OUT_DIM * OUT_DIM);
    for (int i = tid; i < TD*TH*TW; i += NTHREADS) {
        int wo = i % TW;
        int t2 = i / TW;
        int ho = t2 % TH;
        int dd = t2 / TH;
        const float* r = &tmpH[((2*dd)*TH + ho)*TW + wo];
        float v = w0*(r[0] + r[4*TH*TW]) + w1*(r[1*TH*TW] + r[3*TH*TW])
                + w2*r[2*TH*TW];
        os[((size_t)(od0 + dd)*OUT_DIM + (oh0 + ho))*OUT_DIM + (ow0 + wo)] = v;
    }
}

extern "C" void kernel_launch(void* const* d_in, const int* in_sizes, int n_in,
                              void* d_out, int out_size, void* d_ws, size_t ws_size,
                              hipStream_t stream)
{
    const float* x = (const float*)d_in[0];
    // d_in[1] (the filter) is the fixed normalized binomial kernel; weights
    // are hardcoded exactly, so it is not read.
    float* out = (float*)d_out;

    // Allow >64 KB dynamic LDS (CDNA5 WGP supports 320 KB per workgroup).
    (void)hipFuncSetAttribute((const void*)blurpool3d_kernel,
                              hipFuncAttributeMaxDynamicSharedMemorySize,
                              LDS_BYTES);

    dim3 grid(OUT_DIM / TW, OUT_DIM / TH, (OUT_DIM / TD) * N_SLICES);
    blurpool3d_kernel<<<grid, NTHREADS, LDS_BYTES, stream>>>(x, out);
}